// GNN_52862457479735
// MI455X (gfx1250) — compile-verified
//
#include <hip/hip_runtime.h>
#include <hip/hip_bf16.h>

typedef float v2f __attribute__((ext_vector_type(2)));
typedef float v8f __attribute__((ext_vector_type(8)));

#define NEG_SLOPE 0.2f

// ---------------- monotone float<->uint mapping for atomic segment-max ----------
__device__ __forceinline__ unsigned fkey(float f) {
    unsigned b = __float_as_uint(f);
    return (b & 0x80000000u) ? ~b : (b | 0x80000000u);
}
__device__ __forceinline__ float fdecode(unsigned k) {
    return __uint_as_float((k & 0x80000000u) ? (k & 0x7fffffffu) : ~k);
}
#define FKEY_NEG_INF 0x007FFFFFu   // fkey(-inf)

// ---------------- GEMM: H[N,64] = X[N,64] @ W[64,64] via V_WMMA_F32_16X16X4_F32 --
// grid.x = ceil(N/16) row tiles, 128 threads = 4 waves, wave w -> col tile w.
__global__ __launch_bounds__(128) void gat_gemm64(const float* __restrict__ X,
                                                  const float* __restrict__ W,
                                                  float* __restrict__ H, int N) {
    const int rt   = blockIdx.x;
    const int wave = threadIdx.x >> 5;
    const int lane = threadIdx.x & 31;
    const int l15  = lane & 15;
    const int hi   = lane >> 4;
    const int col0 = wave * 16;

    int row = rt * 16 + l15;
    int rowc = row < N ? row : N - 1;          // clamp: keeps EXEC all-ones for WMMA
    const float* xrow = X + (size_t)rowc * 64;

    v8f acc = {};
#pragma unroll
    for (int k = 0; k < 64; k += 4) {
        const int kk = k + 2 * hi;
        v2f a, b;
        a.x = xrow[kk];
        a.y = xrow[kk + 1];
        b.x = W[(size_t)kk * 64 + col0 + l15];
        b.y = W[(size_t)(kk + 1) * 64 + col0 + l15];
        acc = __builtin_amdgcn_wmma_f32_16x16x4_f32(false, a, false, b,
                                                    (short)0, acc, false, false);
    }
#pragma unroll
    for (int v = 0; v < 8; ++v) {
        int ro = rt * 16 + v + 8 * hi;
        if (ro < N) H[(size_t)ro * 64 + col0 + l15] = acc[v];
    }
}

// ---------------- per-node attention dots: a_src = h.att_s, a_dst = h.att_d -----
// one wave32 per node; 2 floats per lane + shuffle reduce.
__global__ __launch_bounds__(256) void gat_attdots(const float* __restrict__ H,
                                                   const float* __restrict__ atts,
                                                   const float* __restrict__ attd,
                                                   float* __restrict__ a_src,
                                                   float* __restrict__ a_dst, int N) {
    const int node = blockIdx.x * (blockDim.x >> 5) + (threadIdx.x >> 5);
    const int lane = threadIdx.x & 31;
    if (node >= N) return;
    float2 hv = ((const float2*)(H + (size_t)node * 64))[lane];
    float2 sv = ((const float2*)atts)[lane];
    float2 dv = ((const float2*)attd)[lane];
    float ds = hv.x * sv.x + hv.y * sv.y;
    float dd = hv.x * dv.x + hv.y * dv.y;
#pragma unroll
    for (int off = 16; off > 0; off >>= 1) {
        ds += __shfl_down(ds, off, 32);
        dd += __shfl_down(dd, off, 32);
    }
    if (lane == 0) { a_src[node] = ds; a_dst[node] = dd; }
}

// ---------------- c = dot(We[0,:], att_e)  (single wave) ------------------------
__global__ void gat_cdot(const float* __restrict__ We,
                         const float* __restrict__ atte, float* __restrict__ out) {
    const int lane = threadIdx.x & 31;
    float2 a = ((const float2*)We)[lane];
    float2 b = ((const float2*)atte)[lane];
    float v = a.x * b.x + a.y * b.y;
#pragma unroll
    for (int off = 16; off > 0; off >>= 1) v += __shfl_down(v, off, 32);
    if (lane == 0) *out = v;
}

// ---------------- init kernels --------------------------------------------------
__global__ void gat_init_nodes(unsigned* __restrict__ segmax,
                               float* __restrict__ den, int N) {
    int i = blockIdx.x * blockDim.x + threadIdx.x;
    if (i < N) { segmax[i] = FKEY_NEG_INF; den[i] = 0.0f; }
}
__global__ void gat_init_out(float* __restrict__ out, const float* __restrict__ b, int n) {
    int i = blockIdx.x * blockDim.x + threadIdx.x;
    if (i < n) out[i] = b[i & 63];
}
__global__ void gat_relu(float* __restrict__ x, int n) {
    int i = blockIdx.x * blockDim.x + threadIdx.x;
    if (i < n) x[i] = fmaxf(x[i], 0.0f);
}

// ---------------- edge logits + segment max (atomic u32 max on monotone key) ----
__global__ __launch_bounds__(256) void gat_edge_logits(const long long* __restrict__ ei,
                                                       const float* __restrict__ ea,
                                                       const float* __restrict__ a_src,
                                                       const float* __restrict__ a_dst,
                                                       const float* __restrict__ cptr,
                                                       float* __restrict__ logits,
                                                       unsigned* __restrict__ segmax,
                                                       int E) {
    int e = blockIdx.x * blockDim.x + threadIdx.x;
    if (e >= E) return;
    int s = (int)ei[e];
    int d = (int)ei[(size_t)E + e];
    float z = a_src[s] + a_dst[d] + (*cptr) * ea[e];
    z = z > 0.0f ? z : NEG_SLOPE * z;          // leaky_relu
    logits[e] = z;
    atomicMax(&segmax[d], fkey(z));
}

// ---------------- exp(logit - max) + segment sum --------------------------------
__global__ __launch_bounds__(256) void gat_edge_exp(const long long* __restrict__ ei,
                                                    float* __restrict__ logits,
                                                    const unsigned* __restrict__ segmax,
                                                    float* __restrict__ den, int E) {
    int e = blockIdx.x * blockDim.x + threadIdx.x;
    if (e >= E) return;
    int d = (int)ei[(size_t)E + e];
    float m = fdecode(segmax[d]);              // finite: node has >=1 edge
    float num = __expf(logits[e] - m);
    logits[e] = num;                           // overwrite with numerator
    atomicAdd(&den[d], num);
}

// ---------------- aggregation: out[dst] += alpha * h[src] (wave per edge) -------
__global__ __launch_bounds__(256) void gat_aggregate(const long long* __restrict__ ei,
                                                     const float* __restrict__ num,
                                                     const float* __restrict__ den,
                                                     const float* __restrict__ H,
                                                     float* __restrict__ out, int E) {
    int e = blockIdx.x * (blockDim.x >> 5) + (threadIdx.x >> 5);
    const int lane = threadIdx.x & 31;
    if (e >= E) return;
    int s = (int)ei[e];
    int d = (int)ei[(size_t)E + e];
    float alpha = num[e] / (den[d] + 1e-16f);
    float2 hv = ((const float2*)(H + (size_t)s * 64))[lane];
    float* o = out + (size_t)d * 64 + lane * 2;
    atomicAdd(o,     alpha * hv.x);
    atomicAdd(o + 1, alpha * hv.y);
}

// ================================================================================
extern "C" void kernel_launch(void* const* d_in, const int* in_sizes, int n_in,
                              void* d_out, int out_size, void* d_ws, size_t ws_size,
                              hipStream_t stream) {
    const float*     x     = (const float*)d_in[0];
    const long long* ei    = (const long long*)d_in[1];
    const float*     ea    = (const float*)d_in[2];
    const float*     W1    = (const float*)d_in[3];
    const float*     atts1 = (const float*)d_in[4];
    const float*     attd1 = (const float*)d_in[5];
    const float*     We1   = (const float*)d_in[6];
    const float*     atte1 = (const float*)d_in[7];
    const float*     b1    = (const float*)d_in[8];
    const float*     W2    = (const float*)d_in[9];
    const float*     atts2 = (const float*)d_in[10];
    const float*     attd2 = (const float*)d_in[11];
    const float*     We2   = (const float*)d_in[12];
    const float*     atte2 = (const float*)d_in[13];
    const float*     b2    = (const float*)d_in[14];

    const int N = in_sizes[0] / 64;
    const int E = in_sizes[1] / 2;

    // ---- carve workspace ----
    char* w = (char*)d_ws;
    auto alloc = [&](size_t bytes) -> char* {
        char* p = w;
        w += (bytes + 255) & ~(size_t)255;
        return p;
    };
    float*    h      = (float*)alloc((size_t)N * 64 * sizeof(float));
    float*    out1   = (float*)alloc((size_t)N * 64 * sizeof(float));
    float*    a_src  = (float*)alloc((size_t)N * sizeof(float));
    float*    a_dst  = (float*)alloc((size_t)N * sizeof(float));
    unsigned* segmax = (unsigned*)alloc((size_t)N * sizeof(unsigned));
    float*    den    = (float*)alloc((size_t)N * sizeof(float));
    float*    logits = (float*)alloc((size_t)E * sizeof(float));
    float*    cs     = (float*)alloc(2 * sizeof(float));

    const int nTiles   = (N + 15) / 16;
    const int nodeWaves= (N + 7) / 8;          // 8 waves / 256-thread block
    const int nodeBlks = (N + 255) / 256;
    const int featBlks = (N * 64 + 255) / 256;
    const int edgeBlks = (E + 255) / 256;
    const int edgeWaves= (E + 7) / 8;

    gat_cdot<<<1, 32, 0, stream>>>(We1, atte1, cs);
    gat_cdot<<<1, 32, 0, stream>>>(We2, atte2, cs + 1);

    // ---------------- layer 1 ----------------
    gat_gemm64<<<nTiles, 128, 0, stream>>>(x, W1, h, N);
    gat_attdots<<<nodeWaves, 256, 0, stream>>>(h, atts1, attd1, a_src, a_dst, N);
    gat_init_nodes<<<nodeBlks, 256, 0, stream>>>(segmax, den, N);
    gat_init_out<<<featBlks, 256, 0, stream>>>(out1, b1, N * 64);
    gat_edge_logits<<<edgeBlks, 256, 0, stream>>>(ei, ea, a_src, a_dst, cs, logits, segmax, E);
    gat_edge_exp<<<edgeBlks, 256, 0, stream>>>(ei, logits, segmax, den, E);
    gat_aggregate<<<edgeWaves, 256, 0, stream>>>(ei, logits, den, h, out1, E);
    gat_relu<<<featBlks, 256, 0, stream>>>(out1, N * 64);

    // ---------------- layer 2 ----------------
    float* out2 = (float*)d_out;
    gat_gemm64<<<nTiles, 128, 0, stream>>>(out1, W2, h, N);
    gat_attdots<<<nodeWaves, 256, 0, stream>>>(h, atts2, attd2, a_src, a_dst, N);
    gat_init_nodes<<<nodeBlks, 256, 0, stream>>>(segmax, den, N);
    gat_init_out<<<featBlks, 256, 0, stream>>>(out2, b2, N * 64);
    gat_edge_logits<<<edgeBlks, 256, 0, stream>>>(ei, ea, a_src, a_dst, cs + 1, logits, segmax, E);
    gat_edge_exp<<<edgeBlks, 256, 0, stream>>>(ei, logits, segmax, den, E);
    gat_aggregate<<<edgeWaves, 256, 0, stream>>>(ei, logits, den, h, out2, E);
}